// StudentTTailFlow_37434934952138
// MI455X (gfx1250) — compile-verified
//
#include <hip/hip_runtime.h>
#include <math.h>

typedef float v2f __attribute__((ext_vector_type(2)));
typedef float v8f __attribute__((ext_vector_type(8)));

#define NEWTON_IT 12
#define CF_IT 16
#define LOG_SQRT_2PI 0.91893853320467274f

__device__ __forceinline__ float frcp(float x) { return __builtin_amdgcn_rcpf(x); }

// ---------------------------------------------------------------------------
// Lentz continued fraction for the regularized incomplete beta.
// All divisions via v_rcp_f32 (1 ulp) — CF convergence is insensitive to this.
// ---------------------------------------------------------------------------
__device__ __forceinline__ float betacf_fast(float a, float b, float x) {
    const float FPMIN = 1e-30f;
    float qab = a + b, qap = a + 1.0f, qam = a - 1.0f;
    float c = 1.0f;
    float d = 1.0f - qab * x * frcp(qap);
    d = (fabsf(d) < FPMIN) ? FPMIN : d;
    d = frcp(d);
    float h = d;
#pragma unroll
    for (int m = 1; m <= CF_IT; ++m) {
        float fm = (float)m;
        float m2 = 2.0f * fm;
        float aa = fm * (b - fm) * x * frcp((qam + m2) * (a + m2));
        d = 1.0f + aa * d; d = (fabsf(d) < FPMIN) ? FPMIN : d; d = frcp(d);
        c = 1.0f + aa * frcp(c); c = (fabsf(c) < FPMIN) ? FPMIN : c;
        h *= d * c;
        aa = -(a + fm) * (qab + fm) * x * frcp((a + m2) * (qap + m2));
        d = 1.0f + aa * d; d = (fabsf(d) < FPMIN) ? FPMIN : d; d = frcp(d);
        c = 1.0f + aa * frcp(c); c = (fabsf(c) < FPMIN) ? FPMIN : c;
        h *= d * c;
    }
    return h;
}

// log Phi(-t) for t >= 0. This is the Newton target -> keep libm accuracy.
__device__ __forceinline__ float log_ndtr_neg(float t) {
    if (t < 5.0f) {
        return logf(0.5f * erfcf(t * 0.70710678118654752f));
    }
    float r = 1.0f / (t * t);
    float corr = log1pf(-r * (1.0f - 3.0f * r * (1.0f - 5.0f * r)));
    return -0.5f * t * t - LOG_SQRT_2PI - logf(t) + corr;
}

// ---------------------------------------------------------------------------
// Prep kernel: per-feature constants -> d_ws (14 SoA arrays of length F).
//   0: shift   1: scale   2: 1/scale
//   pos side (base 3): logc, acoef, 1/df, 0.5*(df+1), lbeta
//   neg side (base 8): same
//   13: mask code (bit0 = pos mask, bit1 = neg mask) as float
// ---------------------------------------------------------------------------
__global__ void ttf_prep(const float* __restrict__ shift,
                         const float* __restrict__ unc,
                         const float* __restrict__ dfp,
                         const float* __restrict__ dfn,
                         const unsigned char* __restrict__ pm,
                         const unsigned char* __restrict__ nm,
                         float* __restrict__ ws, int F) {
    int f = blockIdx.x * blockDim.x + threadIdx.x;
    if (f >= F) return;
    float us = unc[f];
    float sp = (us > 20.0f) ? us : log1pf(expf(us));
    float sc = 0.001f + sp;
    ws[0 * F + f] = shift[f];
    ws[1 * F + f] = sc;
    ws[2 * F + f] = 1.0f / sc;
    for (int side = 0; side < 2; ++side) {
        float df = side ? dfn[f] : dfp[f];
        float logdf = logf(df);
        float logc = lgammaf(0.5f * (df + 1.0f)) - lgammaf(0.5f * df)
                   - 0.5f * (logdf + 1.1447298858494002f);   // log(pi*df)
        int b = side ? 8 : 3;
        ws[(b + 0) * F + f] = logc;
        ws[(b + 1) * F + f] = logc + 0.5f * (df - 1.0f) * logdf;  // tail-init coef
        ws[(b + 2) * F + f] = 1.0f / df;
        ws[(b + 3) * F + f] = 0.5f * (df + 1.0f);
        ws[(b + 4) * F + f] = -(logc + 0.5f * logdf);             // log Beta(df/2, 1/2)
    }
    ws[13 * F + f] = (float)((pm[f] ? 1 : 0) | (nm[f] ? 2 : 0));
}

// ---------------------------------------------------------------------------
// Main kernel: one block per row. 256 threads x 4 features, coalesced I/O.
// log_det reduction: 256 LDS partials -> wave0 V_WMMA_F32_16X16X4_F32 x4.
// ---------------------------------------------------------------------------
__global__ void __launch_bounds__(256) ttf_main(const float* __restrict__ z,
                                                const float* __restrict__ ws,
                                                float* __restrict__ xout,
                                                float* __restrict__ logdet,
                                                int F) {
    const int row = blockIdx.x;
    const int tid = threadIdx.x;
    const float* __restrict__ zr = z + (size_t)row * F;
    float* __restrict__ xr = xout + (size_t)row * F;
    const float* __restrict__ mcode_a = ws + 13 * (size_t)F;

    float acc = 0.0f;

#pragma unroll 1
    for (int f = tid; f < F; f += 256) {
        if (f + 256 < F) __builtin_prefetch(zr + f + 256, 0, 1);  // global_prefetch_b8
        float zv = zr[f];
        float sh = ws[f];
        float sc = ws[F + f];
        float u = (zv - sh) * ws[2 * F + f];
        bool pos = (u >= 0.0f);
        const float* __restrict__ side = ws + (pos ? 3 : 8) * (size_t)F;
        float logc  = side[f];
        float acoef = side[(size_t)F + f];
        float invdf = side[2 * (size_t)F + f];
        float hdf   = side[3 * (size_t)F + f];
        float lbeta = side[4 * (size_t)F + f];
        int mc = (int)mcode_a[f];
        bool masked = pos ? (mc & 1) : (mc & 2);
        float ua = fabsf(u);

        float xv, contrib;
        if (masked) {
            // identity feature: y = u, log_t = log_phi -> contribution 0
            xv = sh + sc * u;
            contrib = 0.0f;
        } else {
            float lp = log_ndtr_neg(ua);                 // log p, p in (0, 0.5]
            float a = hdf - 0.5f;                        // df/2
            float loga = __logf(a);
            float thresh = (a + 1.0f) * frcp(a + 2.5f);
            // tail-asymptotic init, clamped as in reference
            float s = fmaxf(ua, fminf(__expf((acoef - lp) * invdf), 1.0e6f));
#pragma unroll 1
            for (int it = 0; it < NEWTON_IT; ++it) {
                float ss = s * s;
                float e  = ss * invdf;                   // s^2/df
                float xq = frcp(1.0f + e);               // df/(df+s^2)
                float om = e * xq;                       // 1 - xq, computed stably
                float lg1e = __logf(1.0f + e);           // -log(xq)
                // shared log front factor: a*log(xq) + 0.5*log(om) - logBeta
                float T = 0.5f * __logf(om) - a * lg1e - lbeta;
                bool deep = xq < thresh;
                // single CF evaluation with branch-selected arguments
                float ca = deep ? a    : 0.5f;
                float cb = deep ? 0.5f : a;
                float cx = deep ? xq   : om;
                float h = betacf_fast(ca, cb, cx);
                float logF;
                if (deep) {
                    // deep tail, log space: log(0.5 * I_x(a,1/2))
                    logF = T + __logf(h) - 0.69314718f - loga;
                } else {
                    // near median: cdf = 0.5*(1 - 2*h*exp(T))
                    logF = __logf(fmaxf(0.5f - h * __expf(T), 1e-38f));
                }
                float logpdf = logc - hdf * lg1e;
                s = s + (logF - lp) * __expf(logF - logpdf);
                s = fminf(fmaxf(s, 0.0f), 1.0e7f);
            }
            float y = pos ? s : -s;
            xv = sh + sc * y;
            float logphi = -0.5f * u * u - LOG_SQRT_2PI;
            float logt = logc - hdf * log1pf(s * s * invdf);   // accurate path
            contrib = logphi - logt;
        }
        xr[f] = xv;
        acc += contrib;
    }

    // ---- block reduction of 256 partials via WMMA (sum = A x ones) ----
    __shared__ float part[256];
    part[tid] = acc;
    __syncthreads();

    if (tid < 32) {                     // wave 0, EXEC all ones
        int m  = tid & 15;              // A-matrix row
        int kb = (tid >> 4) * 2;        // A-matrix K base for this half-wave
        v8f cacc = {};
        v2f bones; bones.x = 1.0f; bones.y = 1.0f;   // B[4x16] = all ones
#pragma unroll
        for (int c = 0; c < 4; ++c) {   // 4 tiles of 16x4 = 256 partials
            v2f av;
            av.x = part[c * 64 + (kb + 0) * 16 + m];
            av.y = part[c * 64 + (kb + 1) * 16 + m];
            cacc = __builtin_amdgcn_wmma_f32_16x16x4_f32(
                false, av, false, bones, (short)0, cacc, false, false);
        }
        float t = cacc[0] + cacc[1] + cacc[2] + cacc[3]
                + cacc[4] + cacc[5] + cacc[6] + cacc[7];
        t += __shfl_xor(t, 16, 32);
        if (tid == 0) logdet[row] = t;
    }
}

// ---------------------------------------------------------------------------
extern "C" void kernel_launch(void* const* d_in, const int* in_sizes, int n_in,
                              void* d_out, int out_size, void* d_ws, size_t ws_size,
                              hipStream_t stream) {
    const int BF = in_sizes[0];        // B*F
    const int F  = in_sizes[1];        // F
    const int B  = BF / F;

    const float* z     = (const float*)d_in[0];
    const float* shift = (const float*)d_in[1];
    const float* unc   = (const float*)d_in[2];
    const float* dfp   = (const float*)d_in[3];
    const float* dfn   = (const float*)d_in[4];
    const unsigned char* pm = (const unsigned char*)d_in[5];  // jax bool = 1 byte
    const unsigned char* nm = (const unsigned char*)d_in[6];

    float* ws     = (float*)d_ws;
    float* xout   = (float*)d_out;
    float* logdet = xout + (size_t)BF;

    ttf_prep<<<(F + 255) / 256, 256, 0, stream>>>(shift, unc, dfp, dfn, pm, nm, ws, F);
    ttf_main<<<B, 256, 0, stream>>>(z, ws, xout, logdet, F);
}